// Net_7550552506805
// MI455X (gfx1250) — compile-verified
//
#include <hip/hip_runtime.h>
#include <hip/hip_bf16.h>

typedef float v2f __attribute__((ext_vector_type(2)));
typedef float v8f __attribute__((ext_vector_type(8)));

#define DIM      128
#define N_NODES  100000
#define N_EDGES  800000
#define META     20000
#define BATCH    8192
#define HALF_B   4096

// ---------------------------------------------------------------------------
// Generic f32 WMMA GEMM:  out[M,N] = maybe_relu( X[M,K] @ W[N,K]^T + bias[N]
//                                               (+ addend[M,N]) )
// block = (32, 8): 8 waves, each wave computes one 16x16 output tile using
// V_WMMA_F32_16X16X4_F32. Requires M%16==0, N%16==0, K%4==0 (all true here).
// ---------------------------------------------------------------------------
__global__ __launch_bounds__(256)
void gemm_wmma_f32(const float* __restrict__ X, const float* __restrict__ W,
                   const float* __restrict__ bias, const float* __restrict__ addend,
                   float* __restrict__ out, int M, int N, int K, int ldc, int relu)
{
    const int lane = threadIdx.x;            // 0..31
    const int wave = threadIdx.y;            // 0..7
    const int l    = lane & 15;
    const int half = lane >> 4;
    const int m0   = blockIdx.x * 16;
    const int n0   = (blockIdx.y * 8 + wave) * 16;
    if (n0 >= N) return;                     // wave-uniform exit

    const float* xrow = X + (size_t)(m0 + l) * K;
    const float* wrow = W + (size_t)(n0 + l) * K;

    v8f c = {0.f, 0.f, 0.f, 0.f, 0.f, 0.f, 0.f, 0.f};
    for (int k0 = 0; k0 < K; k0 += 4) {
        v2f a = *(const v2f*)(xrow + k0 + 2 * half);   // A: M=l, K=k0+2*half+{0,1}
        v2f b = *(const v2f*)(wrow + k0 + 2 * half);   // B: N=l, K=k0+2*half+{0,1}
        c = __builtin_amdgcn_wmma_f32_16x16x4_f32(
                /*neg_a=*/false, a, /*neg_b=*/false, b,
                /*c_mod=*/(short)0, c, /*reuse_a=*/false, /*reuse_b=*/false);
    }

    const float bn = bias[n0 + l];
    #pragma unroll
    for (int r = 0; r < 8; ++r) {
        const int m = m0 + r + 8 * half;
        float v = c[r] + bn;
        if (addend) v += addend[(size_t)m * N + (n0 + l)];
        if (relu)   v = fmaxf(v, 0.f);
        out[(size_t)m * ldc + (n0 + l)] = v;
    }
}

// theta[i] = dot(emb[i,:], hp[:,0]) ; one wave per node
__global__ __launch_bounds__(256)
void theta_kernel(const float* __restrict__ emb, const float* __restrict__ hp,
                  float* __restrict__ theta, int n_nodes)
{
    const int node = blockIdx.x * 8 + (threadIdx.x >> 5);
    const int lane = threadIdx.x & 31;
    if (node >= n_nodes) return;
    const float* row = emb + (size_t)node * DIM;
    float s = 0.f;
    #pragma unroll
    for (int t = 0; t < 4; ++t) s += row[lane + 32 * t] * hp[lane + 32 * t];
    #pragma unroll
    for (int off = 16; off > 0; off >>= 1) s += __shfl_down(s, off, 32);
    if (lane == 0) theta[node] = s;
}

// one wave per edge: agg[row,:] += exp(interval * theta[col]) * emb[col,:]
__global__ __launch_bounds__(256)
void hawkes_edge_kernel(const float* __restrict__ interval,
                        const int*   __restrict__ edge_index,  // [2, E]
                        const float* __restrict__ emb,         // [N, 128]
                        const float* __restrict__ theta,       // [N]
                        float* __restrict__ agg, int E)
{
    const int wid  = blockIdx.x * 8 + (threadIdx.x >> 5);
    const int lane = threadIdx.x & 31;
    if (wid >= E) return;
    const int row = edge_index[wid];
    const int col = edge_index[E + wid];
    const float decay = expf(interval[wid] * theta[col]);
    const float4 v = *(const float4*)(emb + (size_t)col * DIM + lane * 4);
    float* dst = agg + (size_t)row * DIM + lane * 4;
    atomicAdd(dst + 0, decay * v.x);
    atomicAdd(dst + 1, decay * v.y);
    atomicAdd(dst + 2, decay * v.z);
    atomicAdd(dst + 3, decay * v.w);
}

// outb[i,:] = relu(agg[meta + data[i], :])
__global__ __launch_bounds__(128)
void gather_relu_kernel(const float* __restrict__ agg, const int* __restrict__ data,
                        float* __restrict__ outb, int meta)
{
    const int i = blockIdx.x;
    const int j = threadIdx.x;
    const float v = agg[(size_t)(meta + data[i]) * DIM + j];
    outb[(size_t)i * DIM + j] = fmaxf(v, 0.f);
}

// d[i,:] = |emb[i,:] - emb[i + HALF_B, :]|
__global__ __launch_bounds__(128)
void absdiff_kernel(const float* __restrict__ emb, float* __restrict__ dout)
{
    const int i = blockIdx.x;
    const int j = threadIdx.x;
    dout[(size_t)i * DIM + j] =
        fabsf(emb[(size_t)i * DIM + j] - emb[(size_t)(i + HALF_B) * DIM + j]);
}

// out[i, o] = h[i,:] . W2[o,:] + b2[o]   (M=4096, K=640, N=2)
__global__ __launch_bounds__(256)
void final_kernel(const float* __restrict__ h, const float* __restrict__ W2,
                  const float* __restrict__ b2, float* __restrict__ out)
{
    const int idx = blockIdx.x * blockDim.x + threadIdx.x;
    if (idx >= HALF_B * 2) return;
    const int i = idx >> 1, o = idx & 1;
    const float* hr = h  + (size_t)i * 640;
    const float* wr = W2 + (size_t)o * 640;
    float s = b2[o];
    for (int k = 0; k < 640; ++k) s += hr[k] * wr[k];
    out[idx] = s;
}

extern "C" void kernel_launch(void* const* d_in, const int* in_sizes, int n_in,
                              void* d_out, int out_size, void* d_ws, size_t ws_size,
                              hipStream_t stream)
{
    const float* sem0   = (const float*)d_in[0];
    const float* intv0  = (const float*)d_in[1];
    const int*   eidx0  = (const int*)  d_in[2];
    const float* sem1   = (const float*)d_in[3];
    const float* intv1  = (const float*)d_in[4];
    const int*   eidx1  = (const int*)  d_in[5];
    const float* cen0   = (const float*)d_in[6];
    const float* cen1   = (const float*)d_in[7];
    const int*   data   = (const int*)  d_in[8];
    const float* strct  = (const float*)d_in[9];
    const float* W1 = (const float*)d_in[10]; const float* b1 = (const float*)d_in[11];
    const float* W2 = (const float*)d_in[12]; const float* b2 = (const float*)d_in[13];
    const float* W3 = (const float*)d_in[14]; const float* b3 = (const float*)d_in[15];
    const float* W4 = (const float*)d_in[16]; const float* b4 = (const float*)d_in[17];
    const float* W6 = (const float*)d_in[18]; const float* b6 = (const float*)d_in[19];
    const float* W7 = (const float*)d_in[20]; const float* b7 = (const float*)d_in[21];
    const float* W9 = (const float*)d_in[22]; const float* b9 = (const float*)d_in[23];
    const float* W10 = (const float*)d_in[24]; const float* b10 = (const float*)d_in[25];
    const float* W12 = (const float*)d_in[26]; const float* b12 = (const float*)d_in[27];
    const float* Wc  = (const float*)d_in[28]; const float* bc  = (const float*)d_in[29];
    const float* hp0 = (const float*)d_in[30]; const float* hp1 = (const float*)d_in[31];
    float* out = (float*)d_out;

    // workspace layout (floats)
    float* ws = (float*)d_ws;
    const size_t NE = (size_t)N_NODES * DIM;            // 12,800,000
    float* e_full = ws;                                  // [N_NODES,128]
    float* agg    = e_full + NE;                         // [N_NODES,128]
    float* theta  = agg + NE;                            // [N_NODES]
    float* gath   = theta + 100096;                      // [B,128]
    float* eb0    = gath + (size_t)BATCH * DIM;          // [B,128]
    float* eb1    = eb0  + (size_t)BATCH * DIM;          // [B,128]
    float* cat    = eb1  + (size_t)BATCH * DIM;          // [B,384]
    float* embb   = cat  + (size_t)BATCH * 384;          // [B,128]
    float* dabs   = embb + (size_t)BATCH * DIM;          // [B/2,128]
    float* hbuf   = dabs + (size_t)HALF_B * DIM;         // [B/2,640]

    const dim3 gblk(32, 8, 1);
    #define GEMM(X, W, B_, ADD, OUT, M, N, K, LDC, RELU)                              \
        gemm_wmma_f32<<<dim3((M)/16, ((N)/16 + 7)/8), gblk, 0, stream>>>(             \
            X, W, B_, ADD, OUT, M, N, K, LDC, RELU)

    const float* sems[2]  = {sem0, sem1};
    const float* intvs[2] = {intv0, intv1};
    const int*   eidxs[2] = {eidx0, eidx1};
    const float* Wn[2]    = {W3, W4};   const float* bn[2] = {b3, b4};
    const float* Wg[2]    = {W6, W7};   const float* bg[2] = {b6, b7};
    const float* hps[2]   = {hp0, hp1};
    float* ebs[2]         = {eb0, eb1};

    for (int g = 0; g < 2; ++g) {
        // e = sem @ Wn^T + bn
        GEMM(sems[g], Wn[g], bn[g], nullptr, e_full, N_NODES, DIM, DIM, DIM, 0);
        // theta = e @ hp
        theta_kernel<<<N_NODES / 8, 256, 0, stream>>>(e_full, hps[g], theta, N_NODES);
        // agg = 0
        hipMemsetAsync(agg, 0, NE * sizeof(float), stream);
        // scatter: agg[row] += exp(intv * theta[col]) * e[col]
        hawkes_edge_kernel<<<N_EDGES / 8, 256, 0, stream>>>(
            intvs[g], eidxs[g], e_full, theta, agg, N_EDGES);
        // gath = relu(agg[META + data])
        gather_relu_kernel<<<BATCH, 128, 0, stream>>>(agg, data, gath, META);
        // eb = gath @ Wg^T + bg
        GEMM(gath, Wg[g], bg[g], nullptr, ebs[g], BATCH, DIM, DIM, DIM, 0);
    }

    // cat[:,0:128]   = cen0 @ W9^T  + b9  + eb0
    GEMM(cen0, W9,  b9,  eb0, cat + 0,   BATCH, DIM, DIM, 384, 0);
    // cat[:,128:256] = cen1 @ W10^T + b10 + eb1
    GEMM(cen1, W10, b10, eb1, cat + 128, BATCH, DIM, DIM, 384, 0);
    // cat[:,256:384] = strct @ W12^T + b12
    GEMM(strct, W12, b12, nullptr, cat + 256, BATCH, DIM, DIM, 384, 0);
    // embb = relu(cat @ Wc^T + bc)   (K = 384)
    GEMM(cat, Wc, bc, nullptr, embb, BATCH, DIM, 384, DIM, 1);
    // dabs = |pre - suf|
    absdiff_kernel<<<HALF_B, 128, 0, stream>>>(embb, dabs);
    // hbuf = relu(dabs @ W1^T + b1)  (N = 640)
    GEMM(dabs, W1, b1, nullptr, hbuf, HALF_B, 640, DIM, 640, 1);
    // out = hbuf @ W2^T + b2         (N = 2, scalar path)
    final_kernel<<<(HALF_B * 2 + 255) / 256, 256, 0, stream>>>(hbuf, W2, b2, out);

    #undef GEMM
    (void)in_sizes; (void)n_in; (void)out_size; (void)ws_size;
}